// PEPS_15238543966323
// MI455X (gfx1250) — compile-verified
//
#include <hip/hip_runtime.h>
#include <hip/hip_bf16.h>
#include <stdint.h>

// ---------------------------------------------------------------------------
// PEPS boundary-MPS contraction for MI455X (gfx1250), single persistent
// workgroup (8 wave32 waves). All GEMMs via V_WMMA_F32_16X16X4_F32.
// MPO tensors staged global->LDS with async-to-LDS ops (ASYNCcnt).
// All tensor dims are powers of two -> index math is shift/mask only.
// Workspace requirement: ~9 MB (see WS_FLOATS).
// ---------------------------------------------------------------------------

typedef __attribute__((ext_vector_type(2))) float v2f;
typedef __attribute__((ext_vector_type(8))) float v8f;

#define WG 256
#define NWAVE 8
#define NR 10
#define NC 10
#define DD 8
#define DCAP 64

#define SITE 32768              // 64*8*64 site capacity (floats)
#define OFF_MPS 0
#define OFF_RES (OFF_MPS + NC*SITE)
#define OFF_REN (OFF_RES + NC*SITE)
#define OFF_LEN (OFF_REN + NC*SITE)
#define OFF_MPO (OFF_LEN + NC*SITE)
#define OFF_X1  (OFF_MPO + NC*4096)
#define OFF_X2  (OFF_X1 + 262144)
#define OFF_X3  (OFF_X2 + 262144)
#define OFF_RB  (OFF_X3 + 262144)
#define OFF_QT  (OFF_RB + 32768)
#define OFF_CAR (OFF_QT + 32768)
#define WS_FLOATS (OFF_CAR + 32768)   // ~2.24M floats = ~9 MB

__device__ __forceinline__ int imin(int a, int b) { return a < b ? a : b; }
__device__ __forceinline__ int ctz(int x) { return __builtin_ctz((unsigned)x); }

// ---------------- workgroup-wide helpers (all end in a barrier) ------------

__device__ void wg_copy(const float* in, float* out, int n) {
  for (int t = threadIdx.x; t < n; t += WG) out[t] = in[t];
  __syncthreads();
}

__device__ void wg_setone(float* p) {
  if (threadIdx.x == 0) p[0] = 1.f;
  __syncthreads();
}

// out = permutation of 5-D tensor `in`; every dim is a power of two.
__device__ void perm5(const float* __restrict__ in, float* __restrict__ out,
                      int d0, int d1, int d2, int d3, int d4,
                      int p0, int p1, int p2, int p3, int p4) {
  const int d[5] = {d0, d1, d2, d3, d4};
  const int p[5] = {p0, p1, p2, p3, p4};
  int sh[5];
#pragma unroll
  for (int i = 0; i < 5; i++) sh[i] = ctz(d[i]);
  int ish[5];                       // input strides (shift amounts)
  ish[4] = 0;
  ish[3] = sh[4];
  ish[2] = sh[3] + ish[3];
  ish[1] = sh[2] + ish[2];
  ish[0] = sh[1] + ish[1];
  const int osh1 = sh[p[1]], osh2 = sh[p[2]], osh3 = sh[p[3]], osh4 = sh[p[4]];
  const int om1 = d[p[1]] - 1, om2 = d[p[2]] - 1, om3 = d[p[3]] - 1, om4 = d[p[4]] - 1;
  const int st0 = ish[p[0]], st1 = ish[p[1]], st2 = ish[p[2]], st3 = ish[p[3]], st4 = ish[p[4]];
  const int total = d0 * d1 * d2 * d3 * d4;
  for (int t = threadIdx.x; t < total; t += WG) {
    int r = t;
    const int o4 = r & om4; r >>= osh4;
    const int o3 = r & om3; r >>= osh3;
    const int o2 = r & om2; r >>= osh2;
    const int o1 = r & om1; const int o0 = r >> osh1;
    out[t] = in[(o0 << st0) + (o1 << st1) + (o2 << st2) + (o3 << st3) + (o4 << st4)];
  }
  __syncthreads();
}

// C[M][N] = A op B via V_WMMA_F32_16X16X4_F32.
// A: tA ? (K x M) : (M x K); B: tB ? (N x K) : (K x N). Row-major, tight.
// M, N, K are powers of two. Branchless loads (clamped indices), incremental
// addressing, software-pipelined K loop.
__device__ void gemm(const float* __restrict__ A, const float* __restrict__ B,
                     float* __restrict__ C, int M, int N, int K, bool tA, bool tB) {
  const int lane = threadIdx.x & 31, wave = threadIdx.x >> 5;
  const int half = lane >> 4, mn = lane & 15;
  const int tm = (M + 15) >> 4, tn = (N + 15) >> 4;
  const int tnsh = ctz(tn);                       // tn is a power of two
  const int ntiles = tm << tnsh;
  for (int t = wave; t < ntiles; t += NWAVE) {
    const int m0 = (t >> tnsh) << 4, n0 = (t & (tn - 1)) << 4;
    const int mr = imin(m0 + mn, M - 1);          // clamp: OOB lanes read valid
    const int nc = imin(n0 + mn, N - 1);          // dup data, never stored
    v8f c = {};
    if (K >= 4) {                                 // K multiple of 4: no k guards
      const int aoff = tA ? ((half << 1) * M + mr) : (mr * K + (half << 1));
      const int astp = tA ? (M << 2) : 4;
      const int astr = tA ? M : 1;
      const int boff = tB ? (nc * K + (half << 1)) : ((half << 1) * N + nc);
      const int bstp = tB ? 4 : (N << 2);
      const int bstr = tB ? 1 : N;
      const float* pa = A + aoff;
      const float* pb = B + boff;
      v2f a0, b0;
      a0.x = pa[0]; a0.y = pa[astr];
      b0.x = pb[0]; b0.y = pb[bstr];
      pa += astp; pb += bstp;
      for (int k0 = 4; k0 < K; k0 += 4) {         // pipeline: load next, mma cur
        v2f an, bn;
        an.x = pa[0]; an.y = pa[astr];
        bn.x = pb[0]; bn.y = pb[bstr];
        pa += astp; pb += bstp;
        c = __builtin_amdgcn_wmma_f32_16x16x4_f32(false, a0, false, b0, (short)0, c, false, false);
        a0 = an; b0 = bn;
      }
      c = __builtin_amdgcn_wmma_f32_16x16x4_f32(false, a0, false, b0, (short)0, c, false, false);
    } else {                                      // K in {1,2}: one masked step
      const int ka = half << 1;
      const int kax = imin(ka, K - 1), kay = imin(ka + 1, K - 1);
      const float mx = (ka < K) ? 1.f : 0.f;
      const float my = (ka + 1 < K) ? 1.f : 0.f;
      v2f a, b;
      a.x = mx * A[tA ? kax * M + mr : mr * K + kax];
      a.y = my * A[tA ? kay * M + mr : mr * K + kay];
      b.x = mx * B[tB ? nc * K + kax : kax * N + nc];
      b.y = my * B[tB ? nc * K + kay : kay * N + nc];
      c = __builtin_amdgcn_wmma_f32_16x16x4_f32(false, a, false, b, (short)0, c, false, false);
    }
    if ((m0 + 16 <= M) && (n0 + 16 <= N)) {       // full tile: unguarded stores
      float* pc = C + (m0 + (half << 3)) * N + n0 + mn;
#pragma unroll
      for (int v = 0; v < 8; v++) pc[v * N] = c[v];
    } else {
#pragma unroll
      for (int v = 0; v < 8; v++) {
        const int row = m0 + v + (half << 3);
        if (row < M && (n0 + mn) < N) C[row * N + n0 + mn] = c[v];
      }
    }
  }
  __syncthreads();
}

__device__ float wg_reduce_sum(float v, volatile float* sred) {
  const int lane = threadIdx.x & 31, wv = threadIdx.x >> 5;
  for (int o = 16; o > 0; o >>= 1) v += __shfl_xor(v, o, 32);
  __syncthreads();
  if (lane == 0) sred[wv] = v;
  __syncthreads();
  float s = 0;
#pragma unroll
  for (int i = 0; i < NWAVE; i++) s += sred[i];
  return s;
}

// Modified Gram-Schmidt: A (MxN, row-major, N power of two) -> first k cols
// become Q; R (kxN).
__device__ void mgs_qr(float* A, float* R, int M, int N, int k, volatile float* sred) {
  const int lane = threadIdx.x & 31, wv = threadIdx.x >> 5;
  const int nsh = ctz(N);
  for (int j = 0; j < k; j++) {
    float pv = 0;
    for (int mm = threadIdx.x; mm < M; mm += WG) { float x = A[(mm << nsh) + j]; pv += x * x; }
    const float nrm = sqrtf(wg_reduce_sum(pv, sred));
    const float inv = (nrm > 1e-20f) ? 1.f / nrm : 0.f;
    if (threadIdx.x == 0) R[(j << nsh) + j] = nrm;
    for (int mm = threadIdx.x; mm < M; mm += WG) A[(mm << nsh) + j] *= inv;
    __syncthreads();
    // project column j out of columns j+1..N-1 (one wave per column)
    for (int l = j + 1 + wv; l < N; l += NWAVE) {
      float dt = 0;
      for (int mm = lane; mm < M; mm += 32) dt += A[(mm << nsh) + j] * A[(mm << nsh) + l];
      for (int o = 16; o > 0; o >>= 1) dt += __shfl_xor(dt, o, 32);
      if (lane == 0) R[(j << nsh) + l] = dt;
      for (int mm = lane; mm < M; mm += 32) A[(mm << nsh) + l] -= dt * A[(mm << nsh) + j];
    }
    __syncthreads();
  }
  for (int t = threadIdx.x; t < (k << nsh); t += WG) {
    const int j = t >> nsh, l = t & (N - 1);
    if (l < j) R[t] = 0.f;
  }
  __syncthreads();
}

__device__ void wg_copy_cols(const float* A, int N, float* Q, int M, int k) {
  const int nsh = ctz(N), ksh = ctz(k);
  for (int t = threadIdx.x; t < (M << ksh); t += WG) {
    const int m = t >> ksh, j = t & (k - 1);
    Q[t] = A[(m << nsh) + j];
  }
  __syncthreads();
}

// optimal = einsum('dpr,dla,lwpq,rwb->aqb', m, L, w, R); X1 keeps (r,a,w,q).
__device__ void optimal_step(const float* m, int dm, int P1v, int rm,
                             const float* L, int Ldim, int aDim,
                             const float* w, int Wdim, int P2v,
                             const float* R, int bDim,
                             float* X1, float* X2, float* X3) {
  gemm(m, L, X2, P1v * rm, Ldim * aDim, dm, true, false);           // (p,r,l,a)
  perm5(X2, X3, 1, P1v, rm, Ldim, aDim, 0, 3, 1, 2, 4);             // (l,p,r,a)
  perm5(w, X2, 1, Ldim, Wdim, P1v, P2v, 0, 1, 3, 2, 4);             // (l,p,w,q)
  gemm(X3, X2, X1, rm * aDim, Wdim * P2v, Ldim * P1v, true, false); // (r,a,w,q)  kept
  perm5(X1, X2, 1, rm, aDim, Wdim, P2v, 0, 1, 3, 2, 4);             // (r,w,a,q)
  gemm(X2, R, X3, aDim * P2v, bDim, rm * Wdim, true, false);        // (a,q,b)
}

// Rnew = einsum('aqb,rwb,lwpq,dpr->dla', Ares, Rold, w, m)
__device__ void env_right_step(const float* Ares, int aDim, int P2v, int bDim,
                               const float* Rold, int d2, int W2,
                               const float* w, int L2, int P1v,
                               const float* m, int dDim,
                               float* Rnew, float* X1, float* X2, float* X3) {
  gemm(Ares, Rold, X1, aDim * P2v, d2 * W2, bDim, false, true);     // (a,q,r,w)
  perm5(X1, X2, 1, aDim, P2v, d2, W2, 0, 4, 2, 1, 3);               // (w,q,a,r)
  perm5(w, X3, 1, L2, W2, P1v, P2v, 0, 2, 4, 1, 3);                 // (w,q,l,p)
  gemm(X2, X3, X1, aDim * d2, L2 * P1v, W2 * P2v, true, false);     // (a,r,l,p)
  perm5(X1, X2, 1, aDim, d2, L2, P1v, 0, 4, 2, 3, 1);               // (p,r,l,a)
  gemm(m, X2, Rnew, dDim, L2 * aDim, P1v * d2, false, false);       // (d,l,a)
}

// ---------------------------------------------------------------------------

__global__ __launch_bounds__(WG) void peps_kernel(
    const float* __restrict__ tensors, const int* __restrict__ sample,
    float* __restrict__ out, float* __restrict__ ws) {
  __shared__ float lds_stage[4096];     // 16 KB PEPS-site staging buffer
  __shared__ float sred[NWAVE];
  __shared__ float vbuf[2][DCAP];
  const int lane = threadIdx.x & 31, wave = threadIdx.x >> 5;

  float* MPSb = ws + OFF_MPS; float* RESb = ws + OFF_RES;
  float* RENb = ws + OFF_REN; float* LENb = ws + OFF_LEN;
  float* MPOb = ws + OFF_MPO;
  float* X1 = ws + OFF_X1; float* X2 = ws + OFF_X2; float* X3 = ws + OFF_X3;
  float* RB = ws + OFF_RB; float* QT = ws + OFF_QT; float* CAR = ws + OFF_CAR;

  int mb[NC + 1], rb[NC + 1], erb[NC + 1];
  for (int i = 0; i <= NC; i++) { mb[i] = 1; rb[i] = 1; erb[i] = 1; }
  for (int i = 0; i < NC; i++) wg_setone(MPSb + i * SITE);   // boundary = ones(1,1,1)

  for (int row = 0; row < NR; row++) {
    const int P1 = (row == 0) ? 1 : DD;         // phys (up) of current boundary
    const int P2 = (row == NR - 1) ? 1 : DD;    // phys (down) of new boundary

    // ---- build MPO: async-stage the contiguous 16KB PEPS site into LDS ----
    for (int col = 0; col < NC; col++) {
      const int spin = sample[row * NC + col];
      const float* src = tensors + ((long)(row * NC + col) * 2 + spin) * 4096;
      if (col + 1 < NC)
        __builtin_prefetch(tensors + ((long)(row * NC + col + 1) * 2) * 4096, 0, 1);
      if (wave == 0) {
        unsigned base = (unsigned)(uintptr_t)(&lds_stage[0]);
        for (int it = 0; it < 32; ++it) {                 // 32 iters * 32 lanes * 16B = 16KB
          unsigned goff = (unsigned)((it * 32 + lane) * 16);
          unsigned laddr = base + goff;
          asm volatile("global_load_async_to_lds_b128 %0, %1, %2"
                       :: "v"(laddr), "v"(goff), "s"(src) : "memory");
        }
        asm volatile("s_wait_asynccnt 0" ::: "memory");
      }
      __syncthreads();
      // mpo[l][w][p][q] = t[p][q][l][w], sliced to actual bond dims, tight
      const int L = (col == 0) ? 1 : DD, W = (col == NC - 1) ? 1 : DD;
      const int p2sh = ctz(P2), p1sh = ctz(P1), wsh = ctz(W);
      float* mp = MPOb + col * 4096;
      const int tot = L * W * P1 * P2;
      for (int t = threadIdx.x; t < tot; t += WG) {
        const int q = t & (P2 - 1); int r2 = t >> p2sh;
        const int p = r2 & (P1 - 1); r2 >>= p1sh;
        const int w = r2 & (W - 1); const int l = r2 >> wsh;
        mp[t] = lds_stage[p * 512 + q * 64 + l * 8 + w];
      }
      __syncthreads();
    }

    // ---- init compression (theta + QR, carry R rightward) ----
    int kc = 0, dc = 0, lc = 0;
    rb[0] = 1;
    for (int i = 0; i < NC; i++) {
      const int dm = mb[i], rm = mb[i + 1];
      const int Li = (i == 0) ? 1 : DD, Wi = (i == NC - 1) ? 1 : DD;
      int Mmat, Nmat, leftd;
      if (i == 0) {
        perm5(MPSb, X1, 1, 1, dm, P1, rm, 0, 1, 3, 2, 4);               // (p,d,r)
        perm5(MPOb, X2, 1, Li, Wi, P1, P2, 0, 3, 1, 4, 2);              // (p,l,q,w)
        gemm(X1, X2, X3, dm * rm, Li * P2 * Wi, P1, true, false);       // (d,r,l,q,w)
        perm5(X3, X1, dm, rm, Li, P2, Wi, 0, 2, 3, 1, 4);               // (d,l,q,r,w)
        leftd = dm * Li; Mmat = leftd * P2; Nmat = rm * Wi;
      } else {
        perm5(CAR, X1, 1, 1, kc, dc, lc, 0, 1, 3, 2, 4);                // (d,k,l)
        gemm(MPSb + i * SITE, X1, X2, P1 * rm, kc * lc, dm, true, false); // (p,r,k,l)
        perm5(X2, X1, 1, P1, rm, kc, lc, 0, 4, 1, 3, 2);                // (l,p,k,r)
        perm5(MPOb + i * 4096, X3, 1, Li, Wi, P1, P2, 0, 1, 3, 4, 2);   // (l,p,q,w)
        gemm(X1, X3, X2, kc * rm, P2 * Wi, lc * P1, true, false);       // (k,r,q,w)
        perm5(X2, X1, 1, kc, rm, P2, Wi, 0, 1, 3, 2, 4);                // (k,q,r,w)
        leftd = kc; Mmat = kc * P2; Nmat = rm * Wi;
      }
      if (i == NC - 1) {
        wg_copy(X1, RESb + i * SITE, Mmat * Nmat);
        rb[i] = leftd; rb[i + 1] = Nmat;
      } else {
        const int k = imin(DCAP, imin(Mmat, Nmat));
        mgs_qr(X1, RB, Mmat, Nmat, k, sred);
        wg_copy_cols(X1, Nmat, QT, Mmat, k);
        wg_copy(QT, RESb + i * SITE, Mmat * k);
        wg_copy(RB, CAR, k * Nmat);                                     // carry (k, rm, Wi)
        rb[i] = leftd; rb[i + 1] = k;
        kc = k; dc = rm; lc = Wi;
      }
    }
    for (int i = 0; i <= NC; i++) erb[i] = rb[i];

    // ---- build right environments ----
    wg_setone(RENb + (NC - 1) * SITE);
    for (int i = NC - 2; i >= 0; i--) {
      const int Wi1 = (i + 1 == NC - 1) ? 1 : DD;
      env_right_step(RESb + (i + 1) * SITE, rb[i + 1], P2, rb[i + 2],
                     RENb + (i + 1) * SITE, mb[i + 2], Wi1,
                     MPOb + (i + 1) * 4096, DD, P1,
                     MPSb + (i + 1) * SITE, mb[i + 1],
                     RENb + i * SITE, X1, X2, X3);
    }

    // ---- variational sweeps ----
    for (int sw = 0; sw < 2; sw++) {
      // LR sweep
      wg_setone(LENb);
      for (int i = 0; i < NC; i++) {
        const int dm = mb[i], rm = mb[i + 1];
        const int Li = (i == 0) ? 1 : DD, Wi = (i == NC - 1) ? 1 : DD;
        const int aD = rb[i], bD = erb[i + 1];
        optimal_step(MPSb + i * SITE, dm, P1, rm, LENb + i * SITE, Li, aD,
                     MPOb + i * 4096, Wi, P2, RENb + i * SITE, bD, X1, X2, X3);
        if (i == NC - 1) { wg_copy(X3, RESb + i * SITE, aD * P2 * bD); rb[i + 1] = bD; break; }
        const int Mmat = aD * P2, Nmat = bD;
        const int k = imin(DCAP, imin(Mmat, Nmat));
        mgs_qr(X3, RB, Mmat, Nmat, k, sred);
        wg_copy_cols(X3, Nmat, QT, Mmat, k);
        wg_copy(QT, RESb + i * SITE, Mmat * k);                         // (a,q,k)
        gemm(RB, RESb + (i + 1) * SITE, X2, k, P2 * rb[i + 2], bD, false, false); // absorb C
        wg_copy(X2, RESb + (i + 1) * SITE, k * P2 * rb[i + 2]);
        rb[i + 1] = k;
        perm5(X1, X2, 1, rm, aD, Wi, P2, 0, 2, 4, 1, 3);                // (a,q,r,w)
        gemm(X2, RESb + i * SITE, LENb + (i + 1) * SITE, rm * Wi, k, aD * P2, true, false);
      }
      for (int i = 0; i <= NC; i++) erb[i] = rb[i];

      // RL sweep
      wg_setone(RENb + (NC - 1) * SITE);
      for (int i = NC - 1; i >= 0; i--) {
        const int dm = mb[i], rm = mb[i + 1];
        const int Li = (i == 0) ? 1 : DD, Wi = (i == NC - 1) ? 1 : DD;
        const int aD = erb[i], bD = rb[i + 1];
        optimal_step(MPSb + i * SITE, dm, P1, rm, LENb + i * SITE, Li, aD,
                     MPOb + i * 4096, Wi, P2, RENb + i * SITE, bD, X1, X2, X3);
        if (i == 0) { wg_copy(X3, RESb, aD * P2 * bD); rb[0] = aD; break; }
        perm5(X3, X2, 1, 1, aD, P2, bD, 0, 1, 3, 4, 2);                 // (q,b,a) = mat^T
        const int Mmat = P2 * bD, Nmat = aD;
        const int k = imin(DCAP, imin(Mmat, Nmat));
        mgs_qr(X2, RB, Mmat, Nmat, k, sred);
        wg_copy_cols(X2, Nmat, QT, Mmat, k);                            // (q,b) x k
        perm5(QT, X3, 1, 1, P2, bD, k, 0, 1, 4, 2, 3);                  // (k,q,b)
        wg_copy(X3, RESb + i * SITE, k * P2 * bD);
        gemm(RESb + (i - 1) * SITE, RB, X2, rb[i - 1] * P2, k, aD, false, true); // absorb C^T
        wg_copy(X2, RESb + (i - 1) * SITE, rb[i - 1] * P2 * k);
        rb[i] = k;
        env_right_step(RESb + i * SITE, k, P2, bD,
                       RENb + i * SITE, rm, Wi,
                       MPOb + i * 4096, Li, P1,
                       MPSb + i * SITE, dm,
                       RENb + (i - 1) * SITE, X1, X2, X3);
      }
      for (int i = 0; i <= NC; i++) erb[i] = rb[i];
    }

    // ---- new boundary becomes the MPS for the next row ----
    for (int i = 0; i < NC; i++)
      wg_copy(RESb + i * SITE, MPSb + i * SITE, rb[i] * P2 * rb[i + 1]);
    for (int i = 0; i <= NC; i++) { mb[i] = rb[i]; erb[i] = rb[i]; }
  }

  // ---- collapse final boundary (phys dim = 1) to a scalar ----
  if (threadIdx.x == 0) vbuf[0][0] = 1.f;
  __syncthreads();
  int cur = 0;
  for (int i = 0; i < NC; i++) {
    const float* A = RESb + i * SITE;
    const int a = rb[i], b = rb[i + 1];
    const int bsh = ctz(b);
    for (int bb = threadIdx.x; bb < b; bb += WG) {
      float s = 0;
      for (int aa = 0; aa < a; aa++) s += vbuf[cur][aa] * A[(aa << bsh) + bb];
      vbuf[1 - cur][bb] = s;
    }
    __syncthreads();
    cur ^= 1;
  }
  if (threadIdx.x == 0) out[0] = vbuf[cur][0];
}

extern "C" void kernel_launch(void* const* d_in, const int* in_sizes, int n_in,
                              void* d_out, int out_size, void* d_ws, size_t ws_size,
                              hipStream_t stream) {
  const float* tensors = (const float*)d_in[0];   // (10,10,2,8,8,8,8) f32
  const int* sample = (const int*)d_in[1];        // (100,) i32
  float* out = (float*)d_out;                     // scalar f32
  float* ws = (float*)d_ws;                       // needs >= WS_FLOATS*4 bytes (~9 MB)
  (void)in_sizes; (void)n_in; (void)out_size; (void)ws_size;
  hipLaunchKernelGGL(peps_kernel, dim3(1), dim3(WG), 0, stream,
                     tensors, sample, out, ws);
}